// GlobalMapEncoder_67430986547460
// MI455X (gfx1250) — compile-verified
//
#include <hip/hip_runtime.h>
#include <hip/hip_bf16.h>

#define BB 16
#define TT 32
#define LL 64
#define DD 768
#define GG 256
#define CHUNK 32                 // D-columns per scatter workgroup
#define NCHUNK (DD / CHUNK)      // 24
#define LN_EPS 1e-12f

typedef __attribute__((ext_vector_type(2))) float v2f;
typedef __attribute__((ext_vector_type(4))) float v4f;
typedef __attribute__((ext_vector_type(8))) float v8f;

// ---------------------------------------------------------------------------
// Kernel 1: fused visited-mean + scatter-mean over the 100 MB trajectory
// tensor. grid = (B, D/CHUNK), block = 256. Each WG streams T*L rows x 32
// cols exactly once (non-temporal b128 loads) and accumulates in LDS with
// ds_add_f32 atomics. Visited sums are register-reduced per step first.
// ---------------------------------------------------------------------------
__global__ __launch_bounds__(256) void gmap_img_fts_kernel(
    const float* __restrict__ split,      // [B,T,L,D]
    const int*   __restrict__ vp_lens,    // [B,T]
    const int*   __restrict__ vstep,      // [B,G]
    const int*   __restrict__ cand,       // [B,T,L]
    const int*   __restrict__ glens,      // [B]
    float*       __restrict__ out)        // [B,G,D]
{
    __shared__ float accum[GG * CHUNK];   // 32 KB: unvisited scatter sums
    __shared__ float vis[TT * CHUNK];     //  4 KB: visited (per-step) sums
    __shared__ int   cnt[GG];
    __shared__ int   len_s[TT];

    const int b   = blockIdx.x;
    const int d0  = blockIdx.y * CHUNK;
    const int tid = threadIdx.x;

    for (int i = tid; i < GG * CHUNK; i += 256) accum[i] = 0.f;
    for (int i = tid; i < TT * CHUNK; i += 256) vis[i] = 0.f;
    if (tid < GG) cnt[tid] = 0;
    if (tid < TT) len_s[tid] = vp_lens[b * TT + tid];
    __syncthreads();

    // 8 threads per row, float4 each -> 32 cols; 32 rows in flight.
    const int sub   = tid & 7;            // column slice within the chunk
    const int rbase = tid >> 3;           // row 0..31 within a half-step
    for (int t = 0; t < TT; ++t) {
        const int len = len_s[t];
        float lv0 = 0.f, lv1 = 0.f, lv2 = 0.f, lv3 = 0.f;   // visited partials
#pragma unroll
        for (int half = 0; half < 2; ++half) {
            const int j = half * 32 + rbase;
            const int slot = cand[(b * TT + t) * LL + j];
            const v4f v4 = __builtin_nontemporal_load((const v4f*)(split +
                (((size_t)(b * TT + t)) * LL + j) * DD + d0 + sub * 4));
            const bool inlen = (j < len);
            if (inlen) { lv0 += v4.x; lv1 += v4.y; lv2 += v4.z; lv3 += v4.w; }
            if (slot >= 0) {
                const float m = inlen ? 1.f : 0.f;          // masked values
                float* ap = &accum[slot * CHUNK + sub * 4];
                atomicAdd(ap + 0, v4.x * m);
                atomicAdd(ap + 1, v4.y * m);
                atomicAdd(ap + 2, v4.z * m);
                atomicAdd(ap + 3, v4.w * m);
                if (sub == 0) atomicAdd(&cnt[slot], 1);
            }
        }
        float* vp = &vis[t * CHUNK + sub * 4];
        atomicAdd(vp + 0, lv0);
        atomicAdd(vp + 1, lv1);
        atomicAdd(vp + 2, lv2);
        atomicAdd(vp + 3, lv3);
    }
    __syncthreads();

    // Finalize: per gmap slot pick visited mean vs scatter mean, apply mask.
    const int gl = glens[b];
    for (int e = tid; e < GG * CHUNK; e += 256) {
        const int g = e / CHUNK;
        const int c = e % CHUNK;
        float val = 0.f;
        if (g >= 1 && g < gl) {
            const int vs = vstep[b * GG + g];
            if (vs >= 0)
                val = vis[vs * CHUNK + c] / (float)len_s[vs];
            else
                val = accum[g * CHUNK + c] / fmaxf((float)cnt[g], 1.f);
        }
        out[((size_t)(b * GG + g)) * DD + d0 + c] = val;
    }
}

// ---------------------------------------------------------------------------
// Kernel 2: pos = LN(fts @ W + b) via V_WMMA_F32_16X16X4_F32 (K=7 padded to
// 2 x K=4), then out += step_emb[ids] + pos. One WG per 16 rows of the
// flattened [B*G, D] matrix; 8 waves cover the 48 column tiles.
// ---------------------------------------------------------------------------
__global__ __launch_bounds__(256) void pos_ln_assemble_kernel(
    const float* __restrict__ fts,        // [B*G, 7]
    const float* __restrict__ W,          // [7, D]
    const float* __restrict__ bias,       // [D]
    const float* __restrict__ gamma,      // [D]
    const float* __restrict__ beta,       // [D]
    const float* __restrict__ step_emb,   // [100, D]
    const int*   __restrict__ step_ids,   // [B*G]
    float*       __restrict__ out)        // [B*G, D]
{
    const int STR = DD + 8;               // pad to avoid LDS bank conflicts
    __shared__ float pos_s[16 * (DD + 8)];
    __shared__ float red[2 * 256];
    __shared__ float mu_s[16], rs_s[16];

    const int m0    = blockIdx.x * 16;
    const int tid   = threadIdx.x;
    const int lane  = tid & 31;
    const int wave  = tid >> 5;
    const int idx16 = lane & 15;          // M (for A) / N (for B,C,D) in tile
    const int hi    = lane >> 4;          // lane half selects K pair / M half
    const int m     = m0 + idx16;

    // A fragments (16x4 f32 layout: lanes 0-15 K={0,1}, lanes 16-31 K={2,3})
    v2f a0, a1;
    {
        const float* fp = fts + (size_t)m * 7;
        const int k0 = hi * 2;            // 0 or 2
        a0.x = fp[k0];
        a0.y = fp[k0 + 1];
        const int k1 = 4 + hi * 2;        // 4 or 6
        a1.x = fp[k1];                    // k=4 or 6 (<7, valid)
        a1.y = (k1 + 1 < 7) ? fp[k1 + 1] : 0.f;   // k=5 valid, k=7 padded 0
    }

    for (int nt = wave; nt < DD / 16; nt += 8) {
        const int n0 = nt * 16;
        const int n  = n0 + idx16;
        v2f b0, b1;
        {
            const int k0 = hi * 2;
            b0.x = W[k0 * DD + n];
            b0.y = W[(k0 + 1) * DD + n];
            const int k1 = 4 + hi * 2;
            b1.x = W[k1 * DD + n];
            b1.y = (k1 + 1 < 7) ? W[(k1 + 1) * DD + n] : 0.f;
        }
        v8f c;
        const float bv = bias[n];
#pragma unroll
        for (int v = 0; v < 8; ++v) c[v] = bv;
        c = __builtin_amdgcn_wmma_f32_16x16x4_f32(
                false, a0, false, b0, (short)0, c, false, false);
        c = __builtin_amdgcn_wmma_f32_16x16x4_f32(
                false, a1, false, b1, (short)0, c, false, false);
#pragma unroll
        for (int v = 0; v < 8; ++v) {
            const int row = v + 8 * hi;   // C/D layout: VGPR v, lane half -> M
            pos_s[row * STR + n] = c[v];
        }
    }
    __syncthreads();

    // LayerNorm row reduction: 16 threads per row, 48 contiguous cols each.
    {
        const int row  = tid >> 4;
        const int part = tid & 15;
        float s = 0.f, ss = 0.f;
        const float* rp = &pos_s[row * STR + part * 48];
        for (int i = 0; i < 48; ++i) { const float x = rp[i]; s += x; ss += x * x; }
        red[tid]       = s;
        red[256 + tid] = ss;
    }
    __syncthreads();
    if (tid < 16) {
        float s = 0.f, ss = 0.f;
        for (int i = 0; i < 16; ++i) { s += red[tid * 16 + i]; ss += red[256 + tid * 16 + i]; }
        const float mu  = s / (float)DD;
        const float var = ss / (float)DD - mu * mu;
        mu_s[tid] = mu;
        rs_s[tid] = rsqrtf(var + LN_EPS);
    }
    __syncthreads();

    // out = img_fts (already in out) + step_emb[ids] + LN(pos)
    {
        const int row = tid >> 4;
        const int c0  = (tid & 15) * 48;
        const int mr  = m0 + row;
        const size_t base = (size_t)mr * DD + c0;
        const float* se = step_emb + (size_t)step_ids[mr] * DD + c0;
        const float mu = mu_s[row], rs = rs_s[row];
        for (int i = 0; i < 48; ++i) {
            const float p = (pos_s[row * STR + c0 + i] - mu) * rs * gamma[c0 + i] + beta[c0 + i];
            out[base + i] = out[base + i] + se[i] + p;
        }
    }
}

extern "C" void kernel_launch(void* const* d_in, const int* in_sizes, int n_in,
                              void* d_out, int out_size, void* d_ws, size_t ws_size,
                              hipStream_t stream) {
    const float* split     = (const float*)d_in[0];   // [B,T,L,D]
    const float* pos_fts   = (const float*)d_in[1];   // [B,G,7]
    // d_in[2] txt_embeds: unused (encoder is identity on img branch)
    const float* W_pos     = (const float*)d_in[3];   // [7,D]
    const float* b_pos     = (const float*)d_in[4];   // [D]
    const float* ln_gamma  = (const float*)d_in[5];   // [D]
    const float* ln_beta   = (const float*)d_in[6];   // [D]
    const float* step_emb  = (const float*)d_in[7];   // [100,D]
    const int*   vp_lens   = (const int*)d_in[8];     // [B,T]
    const int*   vstep     = (const int*)d_in[9];     // [B,G]
    const int*   cand      = (const int*)d_in[10];    // [B,T,L]
    const int*   step_ids  = (const int*)d_in[11];    // [B,G]
    const int*   glens     = (const int*)d_in[12];    // [B]
    // d_in[13] txt_masks: unused
    float* out = (float*)d_out;                       // [B,G,D]

    gmap_img_fts_kernel<<<dim3(BB, NCHUNK), 256, 0, stream>>>(
        split, vp_lens, vstep, cand, glens, out);
    pos_ln_assemble_kernel<<<(BB * GG) / 16, 256, 0, stream>>>(
        pos_fts, W_pos, b_pos, ln_gamma, ln_beta, step_emb, step_ids, out);
}